// HarmonicNet_5360119186073
// MI455X (gfx1250) — compile-verified
//
#include <hip/hip_runtime.h>
#include <hip/hip_bf16.h>
#include <math.h>

// ---------------------------------------------------------------------------
// HarmonicNet on gfx1250:
//   edge phase : scatter-add complex harmonic messages (L2-resident atomics)
//   node phase : 3x (Nx16)*(16x16) fp32 GEMMs via v_wmma_f32_16x16x4_f32,
//                fused magnitude + bias + log_softmax epilogue
// Workspace layout (d_ws): aggr[node][plane][k]  plane: 0=re(o0) 1=re(o1)
// 2=im(o1), k = r*8+f  -> 48 floats / node = 19.2 MB (fits L2).
// ---------------------------------------------------------------------------

typedef __attribute__((ext_vector_type(2))) float v2f;
typedef __attribute__((ext_vector_type(8))) float v8f;

#define IN_CH 8
#define NCLS 16
#define AGGR_STRIDE 48   // 3 planes * 16 k-values

__global__ void hn_zero_kernel(float* __restrict__ p, size_t n) {
    size_t i = (size_t)blockIdx.x * blockDim.x + threadIdx.x;
    size_t stride = (size_t)gridDim.x * blockDim.x;
    for (; i < n; i += stride) p[i] = 0.0f;
}

__global__ __launch_bounds__(256)
void hn_edge_kernel(const float* __restrict__ x,
                    const int*   __restrict__ edge_index,
                    const float* __restrict__ precomp,
                    float*       __restrict__ aggr,
                    int E) {
    int e = blockIdx.x * blockDim.x + threadIdx.x;
    if (e >= E) return;

    int s = edge_index[e];       // src (segment id)
    int t = edge_index[E + e];   // tgt (gather id)

    const float* xp = x + (size_t)t * IN_CH;
    float xv[IN_CH];
#pragma unroll
    for (int f = 0; f < IN_CH; ++f) xv[f] = xp[f];

    // precomp[e][o][r][{re,im}]  (O1=2, R=2)
    const float* pc = precomp + (size_t)e * 8;
    float pr00 = pc[0];                  // o=0,r=0 re
    float pr01 = pc[2];                  // o=0,r=1 re
    float pr10 = pc[4];                  // o=1,r=0 re
    float pi10 = pc[5];                  // o=1,r=0 im  (sign[1]=1)
    float pr11 = pc[6];                  // o=1,r=1 re
    float pi11 = pc[7];                  // o=1,r=1 im
    // sign[0]=0 => im(o=0) messages are identically zero: skipped.

    float* base = aggr + (size_t)s * AGGR_STRIDE;
#pragma unroll
    for (int f = 0; f < IN_CH; ++f) {
        float xf = xv[f];
        atomicAdd(base +  0 + f,     xf * pr00);  // plane re(o0), r=0
        atomicAdd(base +  8 + f,     xf * pr01);  // plane re(o0), r=1
        atomicAdd(base + 16 + f,     xf * pr10);  // plane re(o1), r=0
        atomicAdd(base + 24 + f,     xf * pr11);  // plane re(o1), r=1
        atomicAdd(base + 32 + f,     xf * pi10);  // plane im(o1), r=0
        atomicAdd(base + 40 + f,     xf * pi11);  // plane im(o1), r=1
    }
}

// One 16-node tile per wave. 3 GEMM accumulators (re0, re1, im1), each built
// from 4 chained v_wmma_f32_16x16x4_f32 ops (K = 16).
__global__ __launch_bounds__(64)
void hn_node_kernel(const float* __restrict__ aggr,
                    const float* __restrict__ rp,     // radial_profile (4,2,16,8); only o<2 used
                    const float* __restrict__ bias,   // (16,)
                    float*       __restrict__ out,    // (N,16)
                    int nTiles) {
    const int lane = threadIdx.x & 31;
    const int wave = threadIdx.x >> 5;
    const int half = lane >> 4;   // 0: lanes 0-15, 1: lanes 16-31
    const int lid  = lane & 15;

    // B fragments: per kb, VGPR v holds W[k = kb*4 + half*2 + v][c = lid]
    // W_o[k][c] = prof[o][r=k>>3][c][f=k&7]; prof strides: o:256, r:128, c:8, f:1
    v2f w0[4], w1[4];
#pragma unroll
    for (int kb = 0; kb < 4; ++kb) {
#pragma unroll
        for (int v = 0; v < 2; ++v) {
            int k = kb * 4 + half * 2 + v;
            int r = k >> 3, f = k & 7;
            int off = r * 128 + lid * 8 + f;
            w0[kb][v] = rp[off];         // o = 0
            w1[kb][v] = rp[256 + off];   // o = 1
        }
    }
    float bv = bias[lid];

    int tile = blockIdx.x * (blockDim.x >> 5) + wave;
    if (tile >= nTiles) return;   // wave-uniform; live waves keep EXEC all-ones
    int nodeBase = tile * 16;

    // A fragments: lane holds aggr[nodeBase+lid][plane][kb*4 + half*2 + {0,1}]
    const float* ap = aggr + (size_t)(nodeBase + lid) * AGGR_STRIDE;

    v8f dre0 = {}; v8f dre1 = {}; v8f dim1 = {};
#pragma unroll
    for (int kb = 0; kb < 4; ++kb) {
        int ko = kb * 4 + half * 2;
        v2f a0 = *(const v2f*)(ap +  0 + ko);   // re(o0)
        v2f a1 = *(const v2f*)(ap + 16 + ko);   // re(o1)
        v2f a2 = *(const v2f*)(ap + 32 + ko);   // im(o1)
        dre0 = __builtin_amdgcn_wmma_f32_16x16x4_f32(
                   false, a0, false, w0[kb], (short)0, dre0, false, false);
        dre1 = __builtin_amdgcn_wmma_f32_16x16x4_f32(
                   false, a1, false, w1[kb], (short)0, dre1, false, false);
        dim1 = __builtin_amdgcn_wmma_f32_16x16x4_f32(
                   false, a2, false, w1[kb], (short)0, dim1, false, false);
    }

    // Epilogue: D VGPR v -> node = nodeBase + v + 8*half, class = lid.
    // im(o=0) == 0 structurally, so mag0 = sqrt(max(re0^2, eps)).
#pragma unroll
    for (int v = 0; v < 8; ++v) {
        float re0 = dre0[v], re1 = dre1[v], im1 = dim1[v];
        float mag = sqrtf(fmaxf(re0 * re0, 1e-12f)) +
                    sqrtf(fmaxf(re1 * re1 + im1 * im1, 1e-12f));
        float lg = mag + bv;

        // log_softmax over the 16 classes living in this half-wave
        float m = lg;
#pragma unroll
        for (int off = 8; off > 0; off >>= 1)
            m = fmaxf(m, __shfl_xor(m, off, 16));
        float ex = __expf(lg - m);
        float sum = ex;
#pragma unroll
        for (int off = 8; off > 0; off >>= 1)
            sum += __shfl_xor(sum, off, 16);
        float res = lg - m - __logf(sum);

        int node = nodeBase + v + 8 * half;
        out[(size_t)node * NCLS + lid] = res;
    }
}

extern "C" void kernel_launch(void* const* d_in, const int* in_sizes, int n_in,
                              void* d_out, int out_size, void* d_ws, size_t ws_size,
                              hipStream_t stream) {
    const float* x       = (const float*)d_in[0];   // (N, 8)
    const int*   ei      = (const int*)  d_in[1];   // (2, E)
    const float* precomp = (const float*)d_in[2];   // (E, 2, 2, 2)
    // d_in[3] = connection : unused by the reference
    const float* rp      = (const float*)d_in[4];   // (4, 2, 16, 8)
    const float* bias    = (const float*)d_in[5];   // (16,)
    float* out = (float*)d_out;

    const int N = in_sizes[0] / IN_CH;
    const int E = in_sizes[1] / 2;

    float* aggr = (float*)d_ws;
    size_t aggrElems = (size_t)N * AGGR_STRIDE;

    hn_zero_kernel<<<1024, 256, 0, stream>>>(aggr, aggrElems);
    hn_edge_kernel<<<(E + 255) / 256, 256, 0, stream>>>(x, ei, precomp, aggr, E);

    int nTiles = (N + 15) / 16;               // 6250 for N=100000
    int wavesPerBlock = 2;                    // 64-thread blocks
    int blocks = (nTiles + wavesPerBlock - 1) / wavesPerBlock;
    hn_node_kernel<<<blocks, 32 * wavesPerBlock, 0, stream>>>(aggr, rp, bias, out, nTiles);
}